// SmoothSTEMaxMin_33758442947090
// MI455X (gfx1250) — compile-verified
//
#include <hip/hip_runtime.h>
#include <stdint.h>

// out[b,o] = max_i min(x[b,i], w[i,o]);  x:1024x512, w:512x512, out:1024x512 (f32)
// GEMM-shaped tiling over the (min, max) semiring. VALU-bound (5.4e8 f32 ops vs
// 5 MB of traffic), so: register 4x4 tiles + v_max3-friendly reduction, with
// CDNA5 async global->LDS (ASYNCcnt) double-buffered tile staging.

#define B_M 32      // batch rows per block
#define B_N 64      // out cols per block
#define B_K 16      // k (in-dim) per tile
#define XPAD 20     // padded row length for x tile (20*4B = 80B, 16B aligned, 16-bank skew)
#define NTHR 128    // 4 waves

#define M_DIM 1024
#define N_DIM 512
#define K_DIM 512

__global__ __launch_bounds__(NTHR)
void maxmin_mm_kernel(const float* __restrict__ x,
                      const float* __restrict__ w,
                      float* __restrict__ out) {
  __shared__ __align__(16) float xs[2][B_M][XPAD];   // [buf][b-row][k]
  __shared__ __align__(16) float ws[2][B_K][B_N];    // [buf][k][o-col]

  const int tid = threadIdx.x;
  const int tx  = tid & 15;        // col group: 16 groups * 4 cols
  const int ty  = tid >> 4;        // row group: 8 groups * 4 rows
  const int ty4 = ty * 4;
  const int tx4 = tx * 4;
  const int b0  = blockIdx.y * B_M;
  const int o0  = blockIdx.x * B_N;

  // staging decomposition (one b128 for x, two b128 for w, per thread)
  const int xrow  = tid >> 2;      // 0..31
  const int xquad = tid & 3;       // 0..3   (k/4)
  const int wrow  = tid >> 4;      // 0..7   (k), second load uses wrow+8
  const int wquad = tid & 15;      // 0..15  (col/4)

  const float* xg = x + (size_t)(b0 + xrow) * K_DIM + xquad * 4;
  const float* wg = w + (size_t)wrow * N_DIM + o0 + wquad * 4;

  auto issue_tile = [&](int buf, int kt) {
    // per-lane LDS byte offsets (addrspacecast low 32 bits == LDS offset)
    unsigned xl  = (unsigned)(uintptr_t)&xs[buf][xrow][xquad * 4];
    unsigned wl0 = (unsigned)(uintptr_t)&ws[buf][wrow][wquad * 4];
    unsigned wl1 = (unsigned)(uintptr_t)&ws[buf][wrow + 8][wquad * 4];
    uint64_t xa  = (uint64_t)(uintptr_t)(xg + kt * B_K);
    uint64_t wa0 = (uint64_t)(uintptr_t)(wg + (size_t)kt * B_K * N_DIM);
    uint64_t wa1 = wa0 + (size_t)8 * N_DIM * sizeof(float);
    asm volatile("global_load_async_to_lds_b128 %0, %1, off"
                 :: "v"(xl), "v"(xa) : "memory");
    asm volatile("global_load_async_to_lds_b128 %0, %1, off"
                 :: "v"(wl0), "v"(wa0) : "memory");
    asm volatile("global_load_async_to_lds_b128 %0, %1, off"
                 :: "v"(wl1), "v"(wa1) : "memory");
  };

  float acc[4][4];
  #pragma unroll
  for (int mm = 0; mm < 4; ++mm)
    #pragma unroll
    for (int nn = 0; nn < 4; ++nn)
      acc[mm][nn] = -__builtin_inff();

  issue_tile(0, 0);

  const int KT = K_DIM / B_K;   // 32
  for (int kt = 0; kt < KT; ++kt) {
    const int p = kt & 1;
    // my async loads for tile kt (into buf p) are done; barrier makes all
    // waves' tile data visible and proves everyone finished reading buf p^1.
    asm volatile("s_wait_asynccnt 0" ::: "memory");
    __syncthreads();
    if (kt + 1 < KT) issue_tile(p ^ 1, kt + 1);

    #pragma unroll
    for (int k4 = 0; k4 < B_K / 4; ++k4) {
      float xk[4][4];   // [mm][kk]
      #pragma unroll
      for (int mm = 0; mm < 4; ++mm) {
        const float4 t = *(const float4*)&xs[p][ty4 + mm][k4 * 4];
        xk[mm][0] = t.x; xk[mm][1] = t.y; xk[mm][2] = t.z; xk[mm][3] = t.w;
      }
      #pragma unroll
      for (int kp = 0; kp < 2; ++kp) {          // process k in pairs -> v_max3
        const int kb = k4 * 4 + kp * 2;
        const float4 wa4 = *(const float4*)&ws[p][kb][tx4];
        const float4 wb4 = *(const float4*)&ws[p][kb + 1][tx4];
        const float wav[4] = {wa4.x, wa4.y, wa4.z, wa4.w};
        const float wbv[4] = {wb4.x, wb4.y, wb4.z, wb4.w};
        #pragma unroll
        for (int mm = 0; mm < 4; ++mm) {
          const float xa = xk[mm][kp * 2];
          const float xb = xk[mm][kp * 2 + 1];
          #pragma unroll
          for (int nn = 0; nn < 4; ++nn) {
            const float m0 = fminf(xa, wav[nn]);
            const float m1 = fminf(xb, wbv[nn]);
            acc[mm][nn] = fmaxf(fmaxf(acc[mm][nn], m0), m1);  // -> v_max3_num_f32
          }
        }
      }
    }
  }

  float* op = out + (size_t)(b0 + ty4) * N_DIM + o0 + tx4;
  #pragma unroll
  for (int mm = 0; mm < 4; ++mm) {
    *(float4*)(op + (size_t)mm * N_DIM) =
        make_float4(acc[mm][0], acc[mm][1], acc[mm][2], acc[mm][3]);
  }
}

extern "C" void kernel_launch(void* const* d_in, const int* in_sizes, int n_in,
                              void* d_out, int out_size, void* d_ws, size_t ws_size,
                              hipStream_t stream) {
  (void)in_sizes; (void)n_in; (void)d_ws; (void)ws_size; (void)out_size;
  const float* x = (const float*)d_in[0];   // 1024 x 512
  const float* w = (const float*)d_in[1];   // 512 x 512
  float* out = (float*)d_out;               // 1024 x 512
  dim3 grid(N_DIM / B_N, M_DIM / B_M);      // (8, 32) = 256 blocks
  maxmin_mm_kernel<<<grid, NTHR, 0, stream>>>(x, w, out);
}